// GPTBigMLP_53249004536101
// MI455X (gfx1250) — compile-verified
//
#include <hip/hip_runtime.h>

typedef __attribute__((ext_vector_type(2))) float v2f;
typedef __attribute__((ext_vector_type(8))) float v8f;

#define WMMA_F32(a, b, c) \
    __builtin_amdgcn_wmma_f32_16x16x4_f32(false, (a), false, (b), (short)0, (c), false, false)

__device__ __forceinline__ float gelu_tanh(float x) {
    const float c0 = 0.7978845608028654f;   // sqrt(2/pi)
    const float c1 = 0.044715f;
    float x3 = x * x * x;
    return 0.5f * x * (1.0f + tanhf(c0 * (x + c1 * x3)));
}

// CDNA5 async copy: 16 contiguous bytes global -> LDS per lane, tracked by ASYNCcnt.
// INST_OFFSET applies to both the LDS and global addresses (CDNA5 ISA).
template <int BYTE_OFF>
__device__ __forceinline__ void async_copy16(uint32_t lds_off, const float* g) {
    asm volatile("global_load_async_to_lds_b128 %0, %1, off offset:%2"
                 :: "v"(lds_off), "v"(g), "i"(BYTE_OFF) : "memory");
}

template <int N>
__device__ __forceinline__ void wait_asynccnt() {
    asm volatile("s_wait_asynccnt %0" :: "i"(N) : "memory");
}

// ---------------------------------------------------------------------------
// Big GEMM: C[M,Nn] = epilogue(A[M,K] @ B[K,Nn] + bias)
// 256 threads (8 waves), 128x128 tile, BK=16, fp32 WMMA 16x16x4.
// Double-buffered LDS staged with global_load_async_to_lds_b128 (ASYNCcnt).
// EPI = 0: bias + gelu(tanh);  EPI = 1: bias only.
// ---------------------------------------------------------------------------
template <int EPI>
__global__ __launch_bounds__(256) void gemm_tile128(
    const float* __restrict__ A, const float* __restrict__ B,
    const float* __restrict__ bias, float* __restrict__ C,
    int Nn, int K)
{
    constexpr int BM = 128, BN = 128, BK = 16;
    constexpr int APAD = 4;   // row stride 20 floats = 80B: 16B-multiple, conflict-free
    constexpr int BPAD = 4;   // row stride 132 floats = 528B: 16B-multiple
    __shared__ float As[2][BM][BK + APAD];   // row-major: As[buf][m][k]
    __shared__ float Bs[2][BK][BN + BPAD];   // natural:  Bs[buf][k][n]

    const int tid  = threadIdx.x;
    const int lane = tid & 31;
    const int wid  = tid >> 5;          // 0..7
    const int kh   = lane >> 4;         // lane half (0/1)
    const int idx  = lane & 15;

    const int wm = wid & 3;             // 4 row-slabs of 32
    const int wn = wid >> 2;            // 2 col-slabs of 64

    const int blockM = blockIdx.y * BM;
    const int blockN = blockIdx.x * BN;

    // Async staging assignments: each thread moves 32B of A and 32B of B per tile.
    const int aRow  = tid >> 1;         // 0..127
    const int aSegF = (tid & 1) * 8;    // float offset within 16-float row: 0 or 8
    const int bRow  = tid >> 4;         // 0..15
    const int bColF = (tid & 15) * 8;   // float offset within 128-float row

    const float* gA = A + (size_t)(blockM + aRow) * K + aSegF;
    const float* gB = B + (size_t)bRow * Nn + blockN + bColF;

    uint32_t ldsA[2], ldsB[2];
    ldsA[0] = (uint32_t)(uintptr_t)&As[0][aRow][aSegF];
    ldsA[1] = (uint32_t)(uintptr_t)&As[1][aRow][aSegF];
    ldsB[0] = (uint32_t)(uintptr_t)&Bs[0][bRow][bColF];
    ldsB[1] = (uint32_t)(uintptr_t)&Bs[1][bRow][bColF];

    v8f acc[2][4] = {};

    // Prologue: stage tile 0 into buffer 0 (4 async instructions per wave).
    async_copy16<0>(ldsA[0], gA);
    async_copy16<16>(ldsA[0], gA);
    async_copy16<0>(ldsB[0], gB);
    async_copy16<16>(ldsB[0], gB);

    const int T = K / BK;
    for (int i = 0; i < T; ++i) {
        const int cur = i & 1;
        if (i + 1 < T) {
            const int nxt = cur ^ 1;
            const float* nA = gA + (size_t)(i + 1) * BK;
            const float* nB = gB + (size_t)(i + 1) * BK * Nn;
            async_copy16<0>(ldsA[nxt], nA);
            async_copy16<16>(ldsA[nxt], nA);
            async_copy16<0>(ldsB[nxt], nB);
            async_copy16<16>(ldsB[nxt], nB);
            // Async loads complete in order: <=4 outstanding => tile i has landed.
            wait_asynccnt<4>();
        } else {
            wait_asynccnt<0>();
        }
        __syncthreads();

        #pragma unroll
        for (int k4 = 0; k4 < BK; k4 += 4) {
            v2f af[2], bf[4];
            #pragma unroll
            for (int mi = 0; mi < 2; ++mi) {
                const int m = wm * 32 + mi * 16 + idx;
                af[mi] = *(const v2f*)&As[cur][m][k4 + 2 * kh];   // one ds_load_b64
            }
            #pragma unroll
            for (int ni = 0; ni < 4; ++ni) {
                const int n = wn * 64 + ni * 16 + idx;
                bf[ni].x = Bs[cur][k4 + 2 * kh + 0][n];
                bf[ni].y = Bs[cur][k4 + 2 * kh + 1][n];
            }
            #pragma unroll
            for (int mi = 0; mi < 2; ++mi)
                #pragma unroll
                for (int ni = 0; ni < 4; ++ni)
                    acc[mi][ni] = WMMA_F32(af[mi], bf[ni], acc[mi][ni]);
        }
        __syncthreads();   // all waves done reading buf `cur` before it is re-staged
    }

    // Epilogue: D layout -> VGPR v holds row (v + 8*half), col idx.
    #pragma unroll
    for (int ni = 0; ni < 4; ++ni) {
        const int col  = blockN + wn * 64 + ni * 16 + idx;
        const float bc = bias[col];
        #pragma unroll
        for (int mi = 0; mi < 2; ++mi) {
            #pragma unroll
            for (int v = 0; v < 8; ++v) {
                const int row = blockM + wm * 32 + mi * 16 + v + 8 * kh;
                float val = acc[mi][ni][v] + bc;
                if (EPI == 0) val = gelu_tanh(val);
                C[(size_t)row * Nn + col] = val;
            }
        }
    }
}

// ---------------------------------------------------------------------------
// LoRA contraction: S[N,16] = X[N,K] @ W[K,16]. One wave per 16-token tile.
// ---------------------------------------------------------------------------
__global__ __launch_bounds__(256) void lora_contract(
    const float* __restrict__ X, const float* __restrict__ W,
    float* __restrict__ S, int K)
{
    const int tid  = threadIdx.x;
    const int lane = tid & 31;
    const int wid  = tid >> 5;
    const int kh   = lane >> 4;
    const int idx  = lane & 15;
    const int rowBase = blockIdx.x * 128 + wid * 16;

    v8f acc = {};
    const float* xrow = X + (size_t)(rowBase + idx) * K + 2 * kh;
    for (int k0 = 0; k0 < K; k0 += 4) {
        const v2f a = *(const v2f*)(xrow + k0);
        v2f b;
        b.x = W[(k0 + 2 * kh + 0) * 16 + idx];
        b.y = W[(k0 + 2 * kh + 1) * 16 + idx];
        acc = WMMA_F32(a, b, acc);
    }
    #pragma unroll
    for (int v = 0; v < 8; ++v)
        S[(size_t)(rowBase + v + 8 * kh) * 16 + idx] = acc[v];
}

// ---------------------------------------------------------------------------
// LoRA expansion: Out[N,Nn] = S[N,16] @ W[16,Nn]. One wave per 16x16 tile.
// ---------------------------------------------------------------------------
__global__ __launch_bounds__(256) void lora_expand(
    const float* __restrict__ S, const float* __restrict__ W,
    float* __restrict__ Out, int Nn)
{
    const int tid  = threadIdx.x;
    const int lane = tid & 31;
    const int wid  = tid >> 5;
    const int kh   = lane >> 4;
    const int idx  = lane & 15;
    const int rowBase = blockIdx.x * 16;
    const int colBase = blockIdx.y * 128 + wid * 16;

    v8f acc = {};
    #pragma unroll
    for (int k0 = 0; k0 < 16; k0 += 4) {
        const v2f a = *(const v2f*)(S + (size_t)(rowBase + idx) * 16 + k0 + 2 * kh);
        v2f b;
        b.x = W[(size_t)(k0 + 2 * kh + 0) * Nn + colBase + idx];
        b.y = W[(size_t)(k0 + 2 * kh + 1) * Nn + colBase + idx];
        acc = WMMA_F32(a, b, acc);
    }
    #pragma unroll
    for (int v = 0; v < 8; ++v)
        Out[(size_t)(rowBase + v + 8 * kh) * Nn + colBase + idx] = acc[v];
}

// ---------------------------------------------------------------------------
extern "C" void kernel_launch(void* const* d_in, const int* in_sizes, int n_in,
                              void* d_out, int out_size, void* d_ws, size_t ws_size,
                              hipStream_t stream) {
    constexpr int Ntok = 16384, H = 2048, I = 8192, R = 16;

    const float* hidden = (const float*)d_in[0];
    const float* W_fc   = (const float*)d_in[1];
    const float* b_fc   = (const float*)d_in[2];
    const float* W_proj = (const float*)d_in[3];
    const float* b_proj = (const float*)d_in[4];
    const float* Amat   = (const float*)d_in[5];
    const float* Bmat   = (const float*)d_in[6];
    const float* Cmat   = (const float*)d_in[7];
    const float* Dmat   = (const float*)d_in[8];

    float* out = (float*)d_out;                    // [N, H]
    float* ab  = out + (size_t)Ntok * H;           // [N, I]
    float* cd  = ab  + (size_t)Ntok * I;           // [N, H]

    float* h = (float*)d_ws;                       // [N, I]  512 MB
    float* s = h + (size_t)Ntok * I;               // [N, R]
    float* t = s + (size_t)Ntok * R;               // [N, R]

    dim3 blk(256);

    // s = hidden @ A ; ab = s @ B
    lora_contract<<<dim3(Ntok / 128), blk, 0, stream>>>(hidden, Amat, s, H);
    lora_expand<<<dim3(Ntok / 16, I / 128), blk, 0, stream>>>(s, Bmat, ab, I);

    // h = gelu(hidden @ W_fc + b_fc)
    gemm_tile128<0><<<dim3(I / 128, Ntok / 128), blk, 0, stream>>>(hidden, W_fc, b_fc, h, I, H);

    // t = h @ C ; cd = t @ D
    lora_contract<<<dim3(Ntok / 128), blk, 0, stream>>>(h, Cmat, t, I);
    lora_expand<<<dim3(Ntok / 16, H / 128), blk, 0, stream>>>(t, Dmat, cd, H);

    // out = h @ W_proj + b_proj
    gemm_tile128<1><<<dim3(H / 128, Ntok / 128), blk, 0, stream>>>(h, W_proj, b_proj, out, H, I);
}